// SAGEConv_29901562315005
// MI455X (gfx1250) — compile-verified
//
#include <hip/hip_runtime.h>
#include <hip/hip_bf16.h>

typedef __attribute__((ext_vector_type(2))) float v2f;
typedef __attribute__((ext_vector_type(8))) float v8f;

#define IN_FEATS 64
#define OUT_FEATS 64

// ---------------------------------------------------------------------------
// Kernel 1: zero workspace (neigh_sum accumulator + degree counters)
// ---------------------------------------------------------------------------
__global__ void sage_zero_kernel(float* __restrict__ p, int n) {
    int i = blockIdx.x * blockDim.x + threadIdx.x;
    if (i < n) p[i] = 0.0f;
}

// ---------------------------------------------------------------------------
// Kernel 2: edge scatter.  One wave32 per edge; lane l handles features
// 2l, 2l+1 (contiguous float2 of the 64-wide row).  Accumulator lives in L2
// (25.6MB << 192MB) so the f32 atomics stay on-chip.
// ---------------------------------------------------------------------------
__global__ __launch_bounds__(256) void sage_edge_kernel(
    const float* __restrict__ feat, const float* __restrict__ e_feat,
    const int* __restrict__ src, const int* __restrict__ dst,
    float* __restrict__ neigh_sum, float* __restrict__ deg, int n_edges)
{
    int gtid = blockIdx.x * blockDim.x + threadIdx.x;
    int edge = gtid >> 5;
    int lane = threadIdx.x & 31;
    if (edge >= n_edges) return;

    int s = src[edge];
    int d = dst[edge];
    float w = e_feat[edge];

    v2f f = *(const v2f*)(feat + (size_t)s * IN_FEATS + lane * 2);
    float* np = neigh_sum + (size_t)d * IN_FEATS + lane * 2;
    atomicAdd(np + 0, f.x * w);
    atomicAdd(np + 1, f.y * w);
    if (lane == 0) atomicAdd(deg + d, 1.0f);
}

// ---------------------------------------------------------------------------
// Kernel 3: out = feat @ W_self^T + (neigh_sum/deg) @ W_neigh^T + biases,
// via V_WMMA_F32_16X16X4_F32.  One wave per 16x16 output tile; 16 K-steps
// per GEMM, both GEMMs share one accumulator.  Block = 8 waves =
// 2 M-tiles x 4 N-tiles.
//
// A 16x4 f32 layout: lane L -> row (L&15), K = 2*(L>>4)+j in VGPR j.
// B 4x16  layout:    lane L -> col (L&15), same K indexing.
// C/D 16x16 layout:  VGPR v -> M = v + 8*(L>>4), N = (L&15).
// ---------------------------------------------------------------------------
__global__ __launch_bounds__(256) void sage_gemm_kernel(
    const float* __restrict__ feat, const float* __restrict__ neigh_sum,
    const float* __restrict__ deg,
    const float* __restrict__ W_self, const float* __restrict__ b_self,
    const float* __restrict__ W_neigh, const float* __restrict__ b_neigh,
    float* __restrict__ out, int n_nodes)
{
    const int wave = threadIdx.x >> 5;
    const int lane = threadIdx.x & 31;
    const int mtile = blockIdx.x * 2 + (wave >> 2);
    const int ntile = wave & 3;
    const int m0 = mtile * 16;
    const int n0 = ntile * 16;
    if (m0 >= n_nodes) return;   // uniform per-wave; keeps EXEC all-ones

    const int half = lane >> 4;        // 0: lanes 0-15, 1: lanes 16-31
    const int l15  = lane & 15;
    const int arow = m0 + l15;         // A-matrix row owned by this lane
    const int bcol = n0 + l15;         // B-matrix column (output feature)
    const int koff = 2 * half;         // K offset inside each 4-wide step

    v8f c = {};

    // ---- self term: A = feat tile, B = W_self^T ----
    const float* ap = feat + (size_t)arow * IN_FEATS + koff;
    const float* bp = W_self + (size_t)bcol * IN_FEATS + koff;
#pragma unroll
    for (int kk = 0; kk < 16; ++kk) {
        v2f a = *(const v2f*)(ap + kk * 4);
        v2f b = *(const v2f*)(bp + kk * 4);
        c = __builtin_amdgcn_wmma_f32_16x16x4_f32(
                false, a, false, b, (short)0, c, false, false);
    }

    // ---- neighbor term: A = neigh_sum/deg tile, B = W_neigh^T ----
    // Each lane owns a single M-row, so the mean-normalization is a single
    // per-lane scalar folded into the A operand.
    float invd = 1.0f / fmaxf(deg[arow], 1.0f);
    const float* np2 = neigh_sum + (size_t)arow * IN_FEATS + koff;
    const float* bp2 = W_neigh + (size_t)bcol * IN_FEATS + koff;
#pragma unroll
    for (int kk = 0; kk < 16; ++kk) {
        v2f a = *(const v2f*)(np2 + kk * 4);
        a.x *= invd; a.y *= invd;
        v2f b = *(const v2f*)(bp2 + kk * 4);
        c = __builtin_amdgcn_wmma_f32_16x16x4_f32(
                false, a, false, b, (short)0, c, false, false);
    }

    const float bias = b_self[bcol] + b_neigh[bcol];
#pragma unroll
    for (int v = 0; v < 8; ++v) {
        int node = m0 + v + 8 * half;
        out[(size_t)node * OUT_FEATS + bcol] = c[v] + bias;
    }
}

// ---------------------------------------------------------------------------
extern "C" void kernel_launch(void* const* d_in, const int* in_sizes, int n_in,
                              void* d_out, int out_size, void* d_ws, size_t ws_size,
                              hipStream_t stream)
{
    const float* feat    = (const float*)d_in[0];
    const float* e_feat  = (const float*)d_in[1];
    const int*   src     = (const int*)d_in[2];
    const int*   dst     = (const int*)d_in[3];
    const float* W_self  = (const float*)d_in[4];
    const float* b_self  = (const float*)d_in[5];
    const float* W_neigh = (const float*)d_in[6];
    const float* b_neigh = (const float*)d_in[7];
    float* out = (float*)d_out;

    const int n_nodes = in_sizes[0] / IN_FEATS;   // 100000
    const int n_edges = in_sizes[2];              // 1600000

    // workspace layout: [neigh_sum: n_nodes*64 f32][deg: n_nodes f32]
    float* neigh_sum = (float*)d_ws;
    float* deg = neigh_sum + (size_t)n_nodes * IN_FEATS;

    // 1) zero accumulators (contiguous region)
    {
        int n = n_nodes * IN_FEATS + n_nodes;
        int blocks = (n + 255) / 256;
        sage_zero_kernel<<<blocks, 256, 0, stream>>>(neigh_sum, n);
    }

    // 2) edge scatter: one wave32 per edge
    {
        long long threads = (long long)n_edges * 32;
        int blocks = (int)((threads + 255) / 256);
        sage_edge_kernel<<<blocks, 256, 0, stream>>>(
            feat, e_feat, src, dst, neigh_sum, deg, n_edges);
    }

    // 3) WMMA GEMM + bias: 2 M-tiles (32 nodes) per 256-thread block
    {
        int mtiles = (n_nodes + 15) / 16;          // 6250
        int blocks = (mtiles + 1) / 2;             // 3125
        sage_gemm_kernel<<<blocks, 256, 0, stream>>>(
            feat, neigh_sum, deg, W_self, b_self, W_neigh, b_neigh,
            out, n_nodes);
    }
}